// PoseSkeleton_88811333747289
// MI455X (gfx1250) — compile-verified
//
#include <hip/hip_runtime.h>
#include <hip/hip_bf16.h>

// Forward kinematics, 24-joint skeleton, batch 131072. Memory-bound:
// ~390 MB traffic -> ~17us floor at 23.3 TB/s. All output traffic (52% transforms
// + 10% positions) staged in LDS and flushed with coalesced
// global_store_async_from_lds_b128 (ASYNCcnt), s_wait_asynccnt between reuses.
// Inputs: per-lane cached sequential loads (31/32 dword touches hit L0).

constexpr int kJ   = 24;
constexpr int kTPB = 128;                 // threads per block == elements per block
constexpr int kTStride = 132;             // 8 joints * 16 floats + 4 pad words
constexpr int kPStride = 76;              // 72 floats + 4 pad words
constexpr int kParents[kJ] = {0,0,0,0,1,2,3,4,5,6,7,8,9,9,9,12,13,14,16,17,18,19,20,21};

__global__ __launch_bounds__(kTPB)
void fk_pose_kernel(const float* __restrict__ rot,   // [B,24,3,3]
                    const float* __restrict__ pos,   // [B,24,3]
                    float* __restrict__ outT,        // [B,24,4,4]
                    float* __restrict__ outP)        // [B,24,3]
{
    __shared__ float stgT[kTPB * kTStride];          // 67,584 B (8-joint flush group)
    __shared__ float stgP[kTPB * kPStride];          // 38,912 B  -> 106,496 B total, 3 blocks/WGP

    const int tid  = threadIdx.x;
    const int elem = blockIdx.x * kTPB + tid;

    const float* rbase = rot + (size_t)elem * (kJ * 9);
    const float* pbase = pos + (size_t)elem * (kJ * 3);

    // ---- rest positions for this element (18 x b128, lane-sequential) ----
    float P[kJ * 3];
#pragma unroll
    for (int i = 0; i < 18; ++i) {
        float4 t = ((const float4*)pbase)[i];
        P[i*4+0] = t.x; P[i*4+1] = t.y; P[i*4+2] = t.z; P[i*4+3] = t.w;
    }

    // chained transforms (SROA -> registers; <=~5 live simultaneously)
    float TR[kJ][9];
    float TT[kJ][3];

#pragma unroll
    for (int j = 0; j < kJ; ++j) {
        const int q = kParents[j];

        // local rotation rows (dword-aligned b96 loads)
        float L[9];
#pragma unroll
        for (int r = 0; r < 3; ++r) {
            float3 row = *(const float3*)(rbase + j*9 + r*3);
            L[r*3+0] = row.x; L[r*3+1] = row.y; L[r*3+2] = row.z;
        }

        float relx, rely, relz;
        if (j == 0) { relx = P[0]; rely = P[1]; relz = P[2]; }
        else {
            relx = P[j*3+0] - P[q*3+0];
            rely = P[j*3+1] - P[q*3+1];
            relz = P[j*3+2] - P[q*3+2];
        }

        if (j == 0) {
#pragma unroll
            for (int i = 0; i < 9; ++i) TR[0][i] = L[i];
            TT[0][0] = relx; TT[0][1] = rely; TT[0][2] = relz;
        } else {
#pragma unroll
            for (int r = 0; r < 3; ++r) {
                const float a0 = TR[q][r*3+0], a1 = TR[q][r*3+1], a2 = TR[q][r*3+2];
#pragma unroll
                for (int c = 0; c < 3; ++c)
                    TR[j][r*3+c] = a0*L[0*3+c] + a1*L[1*3+c] + a2*L[2*3+c];
                TT[j][r] = TT[q][r] + a0*relx + a1*rely + a2*relz;
            }
        }

        // posed position -> LDS staging (flushed coalesced at the end)
        stgP[tid*kPStride + j*3 + 0] = TT[j][0];
        stgP[tid*kPStride + j*3 + 1] = TT[j][1];
        stgP[tid*kPStride + j*3 + 2] = TT[j][2];

        // joint_transforms row r = [R | t - R*p], row 3 = [0,0,0,1] -> LDS staging
        const float px = P[j*3+0], py = P[j*3+1], pz = P[j*3+2];
        float* s = &stgT[tid * kTStride + (j & 7) * 16];
#pragma unroll
        for (int r = 0; r < 3; ++r) {
            const float r0 = TR[j][r*3+0], r1 = TR[j][r*3+1], r2 = TR[j][r*3+2];
            s[r*4+0] = r0;
            s[r*4+1] = r1;
            s[r*4+2] = r2;
            s[r*4+3] = TT[j][r] - (r0*px + r1*py + r2*pz);
        }
        s[12] = 0.0f; s[13] = 0.0f; s[14] = 0.0f; s[15] = 1.0f;

        // ---- flush 8-joint group: 128 el * 128 words / 4 = 4096 b128 chunks ----
        if ((j & 7) == 7) {
            const int h = j >> 3;                    // compile-time per unrolled iter
            __syncthreads();                         // staging writes visible

            // chunk c = it*128 + tid  ->  e = it*4 + (tid>>5), rr = tid&31 (no division)
            const int w  = tid >> 5;
            const int rr = tid & 31;
            float* g = outT + ((size_t)(blockIdx.x * kTPB + w)) * 384
                            + (size_t)h * 128 + (size_t)rr * 4;
            uint32_t l = (uint32_t)(uintptr_t)&stgT[0]
                       + (uint32_t)((w * kTStride + rr * 4) * 4);
#pragma unroll 4
            for (int it = 0; it < 32; ++it) {
                asm volatile("global_store_async_from_lds_b128 %0, %1, off"
                             :: "v"(g), "v"(l) : "memory");
                g += 4 * 384;                        // +4 elements of output
                l += 4 * kTStride * 4;               // +4 elements of staging (bytes)
            }

            if (h < 2) {
                asm volatile("s_wait_asynccnt 0x0" ::: "memory");  // staging reusable
                __syncthreads();
            } else {
                // ---- final: flush posed positions: 128 el * 72 words / 4 = 2304 chunks ----
                const uint32_t lp0 = (uint32_t)(uintptr_t)&stgP[0];
#pragma unroll 3
                for (int it = 0; it < 18; ++it) {
                    const unsigned c  = (unsigned)(it * kTPB + tid);
                    const unsigned e  = c / 18u;
                    const unsigned r2 = c - e * 18u;
                    float* gp = outP + ((size_t)(blockIdx.x * kTPB + e)) * 72 + r2 * 4;
                    const uint32_t lp = lp0 + (e * kPStride + r2 * 4) * 4;
                    asm volatile("global_store_async_from_lds_b128 %0, %1, off"
                                 :: "v"(gp), "v"(lp) : "memory");
                }
                // no explicit wait: S_ENDPGM performs implicit wait-idle
            }
        }
    }
}

extern "C" void kernel_launch(void* const* d_in, const int* in_sizes, int n_in,
                              void* d_out, int out_size, void* d_ws, size_t ws_size,
                              hipStream_t stream) {
    const float* rot = (const float*)d_in[0];      // joint_rotations [B,24,3,3]
    const float* pos = (const float*)d_in[1];      // joint_positions [B,24,3]
    const int B = in_sizes[0] / (kJ * 9);          // 131072, divisible by kTPB

    float* outT = (float*)d_out;                   // [B,24,4,4]
    float* outP = outT + (size_t)B * kJ * 16;      // [B,24,3]

    dim3 grid(B / kTPB);                           // 1024 blocks
    fk_pose_kernel<<<grid, kTPB, 0, stream>>>(rot, pos, outT, outP);
}